// PlainFCOS_35296041239057
// MI455X (gfx1250) — compile-verified
//
#include <hip/hip_runtime.h>
#include <hip/hip_bf16.h>
#include <math.h>

// ---------------------------------------------------------------------------
// PlainFCOS post-process: top-1000 of sigmoid(cls_pred) over 21M scores,
// then gather + normalize boxes.  Memory-bound threshold-select:
//   pass 1: 64K-bin LDS-privatized histogram of monotone-mapped logits
//   scan  : find threshold bin (suffix count >= K)
//   pass 2: compact candidates (>= threshold bin) -- served mostly from the
//           192MB L2 since cls_pred (84MB) was just streamed through it
//   final : bitonic sort <=4096 candidates in LDS, decode top-K outputs
// sigmoid is monotone -> select on raw logits, apply sigmoid to 1000 values.
// ---------------------------------------------------------------------------

#define NBINS    65536      // 16-bit bins; 256KB LDS (CDNA5: 320KB per WGP)
#define CAND_MAX 4096
#define SORT_N   4096

__device__ __forceinline__ unsigned int fkey(float f) {
  // order-preserving float -> uint map
  unsigned int u = __float_as_uint(f);
  return (u & 0x80000000u) ? ~u : (u | 0x80000000u);
}
__device__ __forceinline__ float fkey_inv(unsigned int m) {
  unsigned int u = (m & 0x80000000u) ? (m & 0x7FFFFFFFu) : ~m;
  return __uint_as_float(u);
}

// ---- zero the global histogram + counters (fresh every call) --------------
__global__ void k_zero(unsigned int* __restrict__ hist,
                       unsigned int* __restrict__ counters) {
  int i = blockIdx.x * blockDim.x + threadIdx.x;
  if (i < NBINS) hist[i] = 0u;
  if (i < 4)     counters[i] = 0u;
}

// ---- pass 1: full 64K-bin histogram, privatized in 256KB LDS --------------
__global__ void k_hist(const float4* __restrict__ cls4, int nv4,
                       unsigned int* __restrict__ hist) {
  extern __shared__ unsigned int lh[];            // NBINS counters
  const int tid = threadIdx.x, bd = blockDim.x;
  for (int i = tid; i < NBINS; i += bd) lh[i] = 0u;
  __syncthreads();

  const int gid = blockIdx.x * bd + tid;
  const int gs  = gridDim.x * bd;
  for (int i = gid; i < nv4; i += gs) {
    __builtin_prefetch(cls4 + i + gs, 0, 0);      // global_prefetch_b8 (gfx1250)
    float4 v = cls4[i];                            // global_load_b128, RT hint:
    atomicAdd(&lh[fkey(v.x) >> 16], 1u);           //   keep cls_pred hot in the
    atomicAdd(&lh[fkey(v.y) >> 16], 1u);           //   192MB L2 for pass 2
    atomicAdd(&lh[fkey(v.z) >> 16], 1u);
    atomicAdd(&lh[fkey(v.w) >> 16], 1u);
  }
  __syncthreads();

  // flush only non-zero bins (Gaussian logits touch only a few K bins)
  for (int i = tid; i < NBINS; i += bd) {
    unsigned int c = lh[i];
    if (c) atomicAdd(&hist[i], c);
  }
}

// ---- find threshold bin T: max T with suffix_count(T) >= K ----------------
__global__ void k_scan(const unsigned int* __restrict__ hist,
                       unsigned int* __restrict__ counters,
                       const int* __restrict__ num_topk) {
  __shared__ unsigned int csum[1024];
  const int tid = threadIdx.x;
  const int CHUNK = NBINS / 1024;                  // 64 bins per thread
  unsigned int s = 0;
  const int base = tid * CHUNK;
  for (int b = 0; b < CHUNK; ++b) s += hist[base + b];
  csum[tid] = s;
  __syncthreads();

  if (tid == 0) {
    const unsigned long long K = (unsigned long long)num_topk[0];
    unsigned long long acc = 0;
    int T = 0;
    for (int c = 1023; c >= 0; --c) {
      if (acc + (unsigned long long)csum[c] >= K) { // crossing inside chunk c
        const int hi = c * CHUNK + CHUNK - 1;
        const int lo = c * CHUNK;
        for (int b = hi; b >= lo; --b) {
          acc += (unsigned long long)hist[b];
          if (acc >= K) { T = b; break; }
        }
        break;
      }
      acc += (unsigned long long)csum[c];
    }
    counters[1] = (unsigned int)T;
  }
}

// ---- pass 2: compact candidates with bin >= T (L2-resident re-read) -------
__global__ void k_compact(const float4* __restrict__ cls4, int nv4,
                          unsigned int* __restrict__ counters,
                          unsigned long long* __restrict__ cand) {
  const unsigned int T = counters[1];
  const int gid = blockIdx.x * blockDim.x + threadIdx.x;
  const int gs  = gridDim.x * blockDim.x;
  for (int i = gid; i < nv4; i += gs) {
    float4 v = cls4[i];
    unsigned int kk[4] = { fkey(v.x), fkey(v.y), fkey(v.z), fkey(v.w) };
#pragma unroll
    for (int c = 0; c < 4; ++c) {
      if ((kk[c] >> 16) >= T) {
        unsigned int pos = atomicAdd(&counters[0], 1u);
        if (pos < CAND_MAX) {
          unsigned int idx = (unsigned int)(i * 4 + c);
          // composite: key high, ~index low -> descending sort gives
          // index-ascending tie-break, matching jax.lax.top_k
          cand[pos] = ((unsigned long long)kk[c] << 32) |
                      (unsigned long long)(0xFFFFFFFFu - idx);
        }
      }
    }
  }
}

// ---- bitonic sort candidates in LDS, decode/write top-K outputs -----------
__global__ void k_sort_finalize(const unsigned long long* __restrict__ cand,
                                const unsigned int* __restrict__ counters,
                                const float4* __restrict__ box_pred,
                                const int* __restrict__ num_classes,
                                const int* __restrict__ num_topk,
                                const int* __restrict__ img_w,
                                const int* __restrict__ img_h,
                                float* __restrict__ out) {
  __shared__ unsigned long long s[SORT_N];        // 32KB
  const int tid = threadIdx.x, bd = blockDim.x;

  unsigned int nc = counters[0];
  if (nc > (unsigned int)CAND_MAX) nc = CAND_MAX;
  for (int i = tid; i < SORT_N; i += bd)
    s[i] = (i < (int)nc) ? cand[i] : 0ull;        // 0 = smallest composite
  __syncthreads();

  // bitonic sort, descending
  for (int k = 2; k <= SORT_N; k <<= 1) {
    for (int j = k >> 1; j > 0; j >>= 1) {
      for (int i = tid; i < SORT_N; i += bd) {
        const int ij = i ^ j;
        if (ij > i) {
          unsigned long long a = s[i], b = s[ij];
          const bool up = ((i & k) == 0);
          const bool sw = up ? (a < b) : (a > b);
          if (sw) { s[i] = b; s[ij] = a; }
        }
      }
      __syncthreads();
    }
  }

  const int K   = num_topk[0];
  const int NCc = num_classes[0];
  const float iw = 1.0f / (float)img_w[0];
  const float ih = 1.0f / (float)img_h[0];

  for (int o = tid; o < K; o += bd) {
    const unsigned long long c = s[o];
    const unsigned int key = (unsigned int)(c >> 32);
    const unsigned int idx = 0xFFFFFFFFu - (unsigned int)(c & 0xFFFFFFFFull);
    const float logit = fkey_inv(key);
    const float score = 1.0f / (1.0f + expf(-logit));
    const int bidx = (int)(idx / (unsigned int)NCc);
    const int lbl  = (int)(idx % (unsigned int)NCc);
    const float4 bb = box_pred[bidx];
    out[o * 4 + 0] = fminf(fmaxf(bb.x * iw, 0.0f), 1.0f);
    out[o * 4 + 1] = fminf(fmaxf(bb.y * ih, 0.0f), 1.0f);
    out[o * 4 + 2] = fminf(fmaxf(bb.z * iw, 0.0f), 1.0f);
    out[o * 4 + 3] = fminf(fmaxf(bb.w * ih, 0.0f), 1.0f);
    out[4 * K + o] = score;
    out[5 * K + o] = (float)lbl;
  }
}

// ---------------------------------------------------------------------------
extern "C" void kernel_launch(void* const* d_in, const int* in_sizes, int n_in,
                              void* d_out, int out_size, void* d_ws, size_t ws_size,
                              hipStream_t stream) {
  const float* cls    = (const float*)d_in[0];
  const float* box    = (const float*)d_in[1];
  const int*   nclass = (const int*)d_in[2];
  const int*   ntopk  = (const int*)d_in[3];
  const int*   imgw   = (const int*)d_in[4];
  const int*   imgh   = (const int*)d_in[5];
  float* out = (float*)d_out;

  const int N   = in_sizes[0];     // 262144 * 80 = 20,971,520 (divisible by 4)
  const int nv4 = N / 4;

  unsigned int* hist     = (unsigned int*)d_ws;
  unsigned int* counters = hist + NBINS;
  unsigned long long* cand =
      (unsigned long long*)((char*)d_ws + (size_t)NBINS * sizeof(unsigned int) + 64);

  // 1) zero histogram + counters
  k_zero<<<(NBINS + 255) / 256, 256, 0, stream>>>(hist, counters);

  // 2) histogram pass: 1024 threads (32 wave32 waves) per WGP, 256KB LDS each
  k_hist<<<512, 1024, NBINS * sizeof(unsigned int), stream>>>(
      (const float4*)cls, nv4, hist);

  // 3) threshold scan (single block)
  k_scan<<<1, 1024, 0, stream>>>(hist, counters, ntopk);

  // 4) compaction pass (reads served largely from 192MB L2)
  k_compact<<<1024, 256, 0, stream>>>((const float4*)cls, nv4, counters, cand);

  // 5) sort + decode + write outputs (single block)
  k_sort_finalize<<<1, 1024, 0, stream>>>(cand, counters, (const float4*)box,
                                          nclass, ntopk, imgw, imgh, out);
}